// CategorySpecificLinear_15642270892654
// MI455X (gfx1250) — compile-verified
//
#include <hip/hip_runtime.h>

#define NUM_CATEGORIES 64
#define IN_DIM  1024
#define OUT_DIM 1024
#define BATCH   32
#define SEQ     512

#define BM 128
#define BN 128
#define BK 32

typedef __attribute__((ext_vector_type(2))) float v2f;
typedef __attribute__((ext_vector_type(8))) float v8f;
typedef __attribute__((ext_vector_type(4))) int   v4i;

#if defined(__gfx1250__) && __has_builtin(__builtin_amdgcn_global_load_async_to_lds_b128)
#define HAVE_ASYNC 1
#else
#define HAVE_ASYNC 0
#endif

__device__ __forceinline__ void async_copy16(const float* g, float* l) {
#if HAVE_ASYNC
    __builtin_amdgcn_global_load_async_to_lds_b128(
        (__attribute__((address_space(1))) v4i*)g,
        (__attribute__((address_space(3))) v4i*)l,
        /*offset=*/0, /*cpol=*/0);
#else
    *(float4*)l = *(const float4*)g;
#endif
}

__device__ __forceinline__ void wait_async() {
#if HAVE_ASYNC
#if __has_builtin(__builtin_amdgcn_s_wait_asynccnt)
    __builtin_amdgcn_s_wait_asynccnt(0);
#else
    asm volatile("s_wait_asynccnt 0" ::: "memory");
#endif
#endif
}

__global__ __launch_bounds__(256)
void cat_linear_wmma_f32(const float* __restrict__ x,
                         const int*   __restrict__ cid,
                         const float* __restrict__ weight,
                         const float* __restrict__ bias,
                         float* __restrict__ out)
{
    __shared__ __align__(16) float As[2][BM * BK];   // A tile: [row 0..127][k 0..31]
    __shared__ __align__(16) float Bs[2][BK * BN];   // B tile: [k 0..31][col 0..127]

    const int b   = blockIdx.z;
    const int m0  = blockIdx.y * BM;
    const int n0  = blockIdx.x * BN;
    const int cat = cid[b];

    const float* xg = x + ((size_t)b * SEQ + m0) * IN_DIM;                 // BM rows, stride IN_DIM
    const float* wg = weight + (size_t)cat * IN_DIM * OUT_DIM + n0;        // K rows,  stride OUT_DIM

    const int tid  = threadIdx.x;
    const int lane = tid & 31;
    const int wave = tid >> 5;
    const int wm   = (wave & 1) * 64;   // wave M offset within block tile
    const int wn   = (wave >> 1) * 32;  // wave N offset within block tile
    const int lm   = lane & 15;         // M/N index within 16x16 fragment
    const int lk   = (lane >> 4) * 2;   // K pair selected by lane half

    v8f acc[4][2];
#pragma unroll
    for (int mi = 0; mi < 4; ++mi)
#pragma unroll
        for (int ni = 0; ni < 2; ++ni)
            acc[mi][ni] = (v8f)0.0f;

    // ---- stage loader: 256 threads x 4 x 16B = 16 KB per tile ----
    auto load_stage = [&](int kt, int buf) {
        const float* xa = xg + kt * BK;
        const float* wa = wg + (size_t)kt * BK * OUT_DIM;
#pragma unroll
        for (int i = 0; i < 4; ++i) {
            int f  = tid + i * 256;            // float4 id, 0..1023
            int ar = f >> 3, ac = (f & 7) * 4; // A: 8 float4 per 32-float row
            async_copy16(xa + (size_t)ar * IN_DIM + ac, &As[buf][f * 4]);
            int br = f >> 5, bc = (f & 31) * 4; // B: 32 float4 per 128-float row
            async_copy16(wa + (size_t)br * OUT_DIM + bc, &Bs[buf][f * 4]);
        }
    };

    // ---- compute one 32-deep K stage: 8 k-steps x 8 WMMAs ----
    auto compute_stage = [&](int buf) {
        const float* A  = As[buf];
        const float* Bm = Bs[buf];
#pragma unroll
        for (int k = 0; k < BK; k += 4) {
            v2f af[4];
            v2f bf[2];
#pragma unroll
            for (int mi = 0; mi < 4; ++mi) {
                // A 16x4 f32 fragment: lane holds A[M=lm][k+lk], A[M=lm][k+lk+1]
                const float* ap = &A[(wm + mi * 16 + lm) * BK + k + lk];
                af[mi] = *(const v2f*)ap;      // 8B aligned: (k+lk) even
            }
#pragma unroll
            for (int ni = 0; ni < 2; ++ni) {
                // B 4x16 f32 fragment: lane holds B[k+lk][N=lm], B[k+lk+1][N=lm]
                v2f t;
                t.x = Bm[(k + lk)     * BN + wn + ni * 16 + lm];
                t.y = Bm[(k + lk + 1) * BN + wn + ni * 16 + lm];
                bf[ni] = t;
            }
#pragma unroll
            for (int mi = 0; mi < 4; ++mi)
#pragma unroll
                for (int ni = 0; ni < 2; ++ni)
                    acc[mi][ni] = __builtin_amdgcn_wmma_f32_16x16x4_f32(
                        /*neg_a=*/false, af[mi],
                        /*neg_b=*/false, bf[ni],
                        /*c_mod=*/(short)0, acc[mi][ni],
                        /*reuse_a=*/false, /*reuse_b=*/false);
        }
    };

    // ---- main loop: double-buffered, async prefetch of next stage ----
    load_stage(0, 0);
    wait_async();
    __syncthreads();

    const int NKT = IN_DIM / BK;  // 32
    for (int kt = 0; kt < NKT; ++kt) {
        const int buf = kt & 1;
        if (kt + 1 < NKT) load_stage(kt + 1, buf ^ 1);
        compute_stage(buf);
        wait_async();
        __syncthreads();
    }

    // ---- epilogue: C layout lanes 0-15 -> M=r, lanes 16-31 -> M=r+8 ----
    float* og = out + ((size_t)b * SEQ + m0) * OUT_DIM + n0;
#pragma unroll
    for (int mi = 0; mi < 4; ++mi) {
#pragma unroll
        for (int ni = 0; ni < 2; ++ni) {
            const int nc = wn + ni * 16 + lm;
            const float bv = bias[(size_t)cat * OUT_DIM + n0 + nc];
#pragma unroll
            for (int r = 0; r < 8; ++r) {
                const int mr = wm + mi * 16 + r + (lane >> 4) * 8;
                og[(size_t)mr * OUT_DIM + nc] = acc[mi][ni][r] + bv;
            }
        }
    }
}

extern "C" void kernel_launch(void* const* d_in, const int* in_sizes, int n_in,
                              void* d_out, int out_size, void* d_ws, size_t ws_size,
                              hipStream_t stream) {
    const float* x      = (const float*)d_in[0];
    const int*   cid    = (const int*)d_in[1];
    const float* weight = (const float*)d_in[2];
    const float* bias   = (const float*)d_in[3];
    float* out          = (float*)d_out;

    dim3 grid(OUT_DIM / BN, SEQ / BM, BATCH);  // (8, 4, 32) = 1024 blocks
    dim3 block(256);                           // 8 wave32 waves
    hipLaunchKernelGGL(cat_linear_wmma_f32, grid, block, 0, stream,
                       x, cid, weight, bias, out);
}